// CRLLoss_22316650070817
// MI455X (gfx1250) — compile-verified
//
#include <hip/hip_runtime.h>

typedef float v2f __attribute__((ext_vector_type(2)));
typedef float v4f __attribute__((ext_vector_type(4)));
typedef float v8f __attribute__((ext_vector_type(8)));

#define WAVES_PER_BLOCK 8
#define BLOCK 256
#define NCHUNK 8           // 8 chunks * 128 elems = up to C=1024
#define CHUNK 128          // 32 lanes * 4 floats

__device__ __forceinline__ float neg_inf() { return -__builtin_inff(); }

// Kernel 1: zero flags table and the two global accumulators (every call).
__global__ void crl_init(float* __restrict__ acc, int* __restrict__ flags, int C) {
    int i = blockIdx.x * blockDim.x + threadIdx.x;
    if (i < C) flags[i] = 0;
    if (i == 0) { acc[0] = 0.0f; acc[1] = 0.0f; }
}

// Kernel 2: mark minority classes.
__global__ void crl_mark(const long long* __restrict__ minc, int M,
                         int* __restrict__ flags, int C) {
    int i = blockIdx.x * blockDim.x + threadIdx.x;
    if (i < M) {
        long long c = minc[i];
        if (c >= 0 && c < C) flags[c] = 1;
    }
}

// Kernel 3: one wave per row. Row cached in VGPRs (single HBM pass),
// max via shuffle tree, sum(exp) wave-reduction via V_WMMA_F32_16X16X4_F32.
__global__ __launch_bounds__(BLOCK) void crl_main(
        const float* __restrict__ x, const long long* __restrict__ label,
        const int* __restrict__ flags, float* __restrict__ acc,
        int N, int C) {
    __shared__ float s_ce[WAVES_PER_BLOCK];
    __shared__ float s_cnt[WAVES_PER_BLOCK];

    const int wave = threadIdx.x >> 5;
    const int lane = threadIdx.x & 31;
    const long long row = (long long)blockIdx.x * WAVES_PER_BLOCK + wave;
    const bool wvalid = row < (long long)N;
    const long long r = wvalid ? row : 0;          // clamp: no wave-level early exit
    const float* __restrict__ rp = x + r * (long long)C;
    const long long lab = label[r];

    // ---- load row into registers (coalesced 512B/wave b128 NT loads) ----
    float v[NCHUNK * 4];
    float xl = 0.0f;                               // row[label] (one lane finds it)
    #pragma unroll
    for (int i = 0; i < NCHUNK; ++i) {
        const int e = i * CHUNK + lane * 4;
        float q0, q1, q2, q3;
        if (e + 3 < C) {
            v4f q = __builtin_nontemporal_load(reinterpret_cast<const v4f*>(rp + e));
            q0 = q[0]; q1 = q[1]; q2 = q[2]; q3 = q[3];
        } else {
            q0 = (e + 0 < C) ? rp[e + 0] : neg_inf();
            q1 = (e + 1 < C) ? rp[e + 1] : neg_inf();
            q2 = (e + 2 < C) ? rp[e + 2] : neg_inf();
            q3 = (e + 3 < C) ? rp[e + 3] : neg_inf();
        }
        v[4 * i + 0] = q0; v[4 * i + 1] = q1;
        v[4 * i + 2] = q2; v[4 * i + 3] = q3;
        if ((long long)(e + 0) == lab) xl = q0;
        if ((long long)(e + 1) == lab) xl = q1;
        if ((long long)(e + 2) == lab) xl = q2;
        if ((long long)(e + 3) == lab) xl = q3;
    }

    // ---- row max: 31 in-lane fmax + 5-step wave32 butterfly ----
    float m = neg_inf();
    #pragma unroll
    for (int i = 0; i < NCHUNK * 4; ++i) m = fmaxf(m, v[i]);
    #pragma unroll
    for (int off = 16; off >= 1; off >>= 1)
        m = fmaxf(m, __shfl_xor(m, off, 32));

    // ---- per-lane partial sum of exp (invalid elems are -inf -> exp = 0) ----
    float p = 0.0f;
    #pragma unroll
    for (int i = 0; i < NCHUNK * 4; ++i) p += __expf(v[i] - m);

    // ---- row[label]: exactly one lane holds a nonzero -> shuffle-sum ----
    #pragma unroll
    for (int off = 16; off >= 1; off >>= 1)
        xl += __shfl_xor(xl, off, 32);

    // ---- wave-wide sum of p on the matrix pipe ----
    // A (16x4 f32, 2 VGPRs/lane): a0 = p, a1 = 0 -> every lane's p appears
    // exactly once in A. B = all ones (layout-independent). Then
    // D[m][n] = sum_k A[m][k], and sum over all D rows == sum over all lanes p.
    // D layout: lane<16 holds rows 0..7 in d[0..7], lane>=16 holds rows 8..15,
    // so in-lane sum of d[0..7] + one cross-half shuffle gives the total.
    v2f a; a[0] = p;    a[1] = 0.0f;
    v2f b; b[0] = 1.0f; b[1] = 1.0f;
    v8f c = {};
    c = __builtin_amdgcn_wmma_f32_16x16x4_f32(
            /*neg_a=*/false, a, /*neg_b=*/false, b,
            /*c_mod=*/(short)0, c, /*reuse_a=*/false, /*reuse_b=*/false);
    float t = ((c[0] + c[1]) + (c[2] + c[3])) + ((c[4] + c[5]) + (c[6] + c[7]));
    const float total = t + __shfl_xor(t, 16, 32);

    // ---- cross entropy for this row ----
    const float ce = __logf(total) + m - xl;
    const int fl = (lab >= 0 && lab < (long long)C) ? flags[(int)lab] : 1;
    const float kf = (wvalid && fl == 0) ? 1.0f : 0.0f;

    if (lane == 0) { s_ce[wave] = ce * kf; s_cnt[wave] = kf; }
    __syncthreads();
    if (threadIdx.x == 0) {
        float sc = 0.0f, sn = 0.0f;
        #pragma unroll
        for (int w = 0; w < WAVES_PER_BLOCK; ++w) { sc += s_ce[w]; sn += s_cnt[w]; }
        atomicAdd(&acc[0], sc);
        atomicAdd(&acc[1], sn);
    }
}

// Kernel 4: finalize -> out = sum_ce / max(count, 1)
__global__ void crl_fin(const float* __restrict__ acc, float* __restrict__ out) {
    if (blockIdx.x == 0 && threadIdx.x == 0)
        out[0] = acc[0] / fmaxf(acc[1], 1.0f);
}

extern "C" void kernel_launch(void* const* d_in, const int* in_sizes, int n_in,
                              void* d_out, int out_size, void* d_ws, size_t ws_size,
                              hipStream_t stream) {
    const float*     cls   = (const float*)d_in[0];      // [N, C] f32
    const long long* label = (const long long*)d_in[1];  // [N] i64
    const long long* minc  = (const long long*)d_in[2];  // [M] i64

    const int NC = in_sizes[0];
    const int N  = in_sizes[1];
    const int M  = in_sizes[2];
    const int C  = NC / N;                               // 1000 (must be <= 1024)

    float* acc  = (float*)d_ws;                          // [0]=sum_ce, [1]=count
    int*   flags = (int*)((float*)d_ws + 2);             // [C] minority flags

    crl_init<<<(C + 255) / 256, 256, 0, stream>>>(acc, flags, C);
    crl_mark<<<(M + 255) / 256, 256, 0, stream>>>(minc, M, flags, C);

    const int nblocks = (N + WAVES_PER_BLOCK - 1) / WAVES_PER_BLOCK;
    crl_main<<<nblocks, BLOCK, 0, stream>>>(cls, label, flags, acc, N, C);

    crl_fin<<<1, 1, 0, stream>>>(acc, (float*)d_out);
}